// CorrelationFilter_16088947490820
// MI455X (gfx1250) — compile-verified
//
#include <hip/hip_runtime.h>

// CDNA5 / gfx1250, wave32. Fused depthwise-xcorr via fp32 WMMA.
typedef float v2f __attribute__((ext_vector_type(2)));
typedef float v8f __attribute__((ext_vector_type(8)));

#define B_N      128
#define C_N      256
#define NPIX     484            // 22*22 pixels per batch image
#define NPIX_PAD 496            // 31 full M-tiles of 16 -> unguarded D stores
#define NTAP     36             // 6*6 exemplar taps
#define ZSTRIDE  260            // 256 + 4 pad -> conflict-free LDS b64 reads
#define ZTAPS    48             // taps padded to 3 N-tiles of 16
#define ZFLOATS  (ZTAPS * ZSTRIDE)            // 12480
#define TSTRIDE  48
#define TFLOATS  (NPIX_PAD * TSTRIDE)         // 23808
#define SMEM_BYTES ((ZFLOATS + TFLOATS) * 4)  // 145152 B < 160 KB -> 2 WGs/WGP

__global__ __launch_bounds__(1024)
void xcorr_wmma_kernel(const float* __restrict__ x,     // [128,22,22,256]
                       const float* __restrict__ z,     // [128,6,6,256]
                       const float* __restrict__ bias,  // [1,17,17,32768]
                       float* __restrict__ out)         // [128,17,17,1]
{
    extern __shared__ float smem[];
    float* z_s = smem;              // [48][260] zero-padded exemplar
    float* t_s = smem + ZFLOATS;    // [496][48] tap-correlation tensor T

    const int b    = blockIdx.x;
    const int tid  = threadIdx.x;
    const int wave = tid >> 5;
    const int lane = tid & 31;
    const int n    = lane & 15;     // WMMA column / A-row selector
    const int hi   = lane >> 4;     // half-wave -> K sub-pair {0,1} vs {2,3}

    // ---- stage 0: cooperative load z[b] into padded, zero-filled LDS ----
    const float* zb = z + (size_t)b * (NTAP * C_N);
    for (int idx = tid; idx < ZFLOATS; idx += 1024) {
        const int tap = idx / ZSTRIDE;
        const int c   = idx - tap * ZSTRIDE;
        z_s[idx] = (tap < NTAP && c < C_N) ? zb[tap * C_N + c] : 0.0f;
    }
    __syncthreads();

    // ---- stage 1: T = X(484x256) * Z^T(256x36) via v_wmma_f32_16x16x4_f32 ----
    // wave w owns pixel M-tile w (31 tiles cover 484 pixels, tail clamped).
    if (wave < 31) {
        const int pix0 = wave * 16;
        const int pixm = pix0 + n;
        const int pixc = pixm < (NPIX - 1) ? pixm : (NPIX - 1);  // clamp, no divergence
        const float* ax = x + (size_t)b * (NPIX * C_N) + (size_t)pixc * C_N + 2 * hi;
        const float* bx = z_s + 2 * hi;

        v8f acc0 = {0.f,0.f,0.f,0.f,0.f,0.f,0.f,0.f};
        v8f acc1 = acc0, acc2 = acc0;

        #pragma unroll 4
        for (int k = 0; k < C_N; k += 4) {
            const v2f a  = *(const v2f*)(ax + k);
            const v2f b0 = *(const v2f*)(bx + (0 * 16 + n) * ZSTRIDE + k);
            const v2f b1 = *(const v2f*)(bx + (1 * 16 + n) * ZSTRIDE + k);
            const v2f b2 = *(const v2f*)(bx + (2 * 16 + n) * ZSTRIDE + k);
            acc0 = __builtin_amdgcn_wmma_f32_16x16x4_f32(false, a, false, b0,
                                                         (short)0, acc0, false, false);
            acc1 = __builtin_amdgcn_wmma_f32_16x16x4_f32(false, a, false, b1,
                                                         (short)0, acc1, false, false);
            acc2 = __builtin_amdgcn_wmma_f32_16x16x4_f32(false, a, false, b2,
                                                         (short)0, acc2, false, false);
        }

        // D layout (§7.12.2): vgpr v, lane -> row pix0 + v + 8*hi, col n.
        // T padded to 496 rows: rows 484..495 hold garbage, never read.
        #pragma unroll
        for (int v = 0; v < 8; ++v) {
            const int row = pix0 + v + 8 * hi;
            t_s[row * TSTRIDE +  0 + n] = acc0[v];
            t_s[row * TSTRIDE + 16 + n] = acc1[v];
            t_s[row * TSTRIDE + 32 + n] = acc2[v];
        }
    }
    __syncthreads();

    // ---- stage 2: out[b,i,j] = sum_{p,q} T[(i+p)*22+(j+q), p*6+q] + sum_c bias ----
    for (int pos = wave; pos < 289; pos += 32) {
        const int i = pos / 17;
        const int j = pos - i * 17;
        float partial = 0.0f;

        // bias channel sum: lanes stride the 256 contiguous channels (coalesced)
        const float* bp = bias + (size_t)pos * (B_N * C_N) + (size_t)b * C_N;
        #pragma unroll
        for (int t = 0; t < 8; ++t) partial += bp[lane + 32 * t];

        // taps 0..31 on lane, taps 32..35 on lanes 0..3
        {
            const int p = lane / 6, q = lane - p * 6;       // lane < 36 always (wave32)
            partial += t_s[((i + p) * 22 + (j + q)) * TSTRIDE + lane];
        }
        if (lane < NTAP - 32) {
            const int tap = lane + 32;
            const int p = tap / 6, q = tap - p * 6;
            partial += t_s[((i + p) * 22 + (j + q)) * TSTRIDE + tap];
        }

        // wave32 butterfly reduction
        #pragma unroll
        for (int m = 16; m >= 1; m >>= 1) partial += __shfl_xor(partial, m, 32);

        if (lane == 0) out[(size_t)b * 289 + pos] = partial;
    }
}

extern "C" void kernel_launch(void* const* d_in, const int* in_sizes, int n_in,
                              void* d_out, int out_size, void* d_ws, size_t ws_size,
                              hipStream_t stream) {
    const float* x    = (const float*)d_in[0];
    const float* z    = (const float*)d_in[1];
    const float* bias = (const float*)d_in[2];
    float* out        = (float*)d_out;

    // opt in to >64KB dynamic LDS (CDNA5 WGP: 320 KB); idempotent, capture-safe
    hipFuncSetAttribute(reinterpret_cast<const void*>(xcorr_wmma_kernel),
                        hipFuncAttributeMaxDynamicSharedMemorySize, SMEM_BYTES);

    xcorr_wmma_kernel<<<dim3(B_N), dim3(1024), SMEM_BYTES, stream>>>(x, z, bias, out);
}